// PerspectiveTransformLayer_37400575213882
// MI455X (gfx1250) — compile-verified
//
#include <hip/hip_runtime.h>
#include <cstdint>

// PerspectiveTransformLayer: fused tile+warp(nearest)+mask+avg
// image:      (B=4, H=64, W=64, C=128) f32
// transforms: (B=4, N=10, 8)           f32
// mask:       (B=4, N=10, H=64, W=64)  f32
// out:        (B, H, W, C)             f32
//
// Wave32 mapping: one output pixel per wave; 32 lanes x float4 = 128 channels.
// All homography math is wave-uniform -> scalarized via readfirstlane.
// Mask values staged through the CDNA5 async global->LDS path (ASYNCcnt),
// overlapped with the homography setup. Row gathers are software-pipelined:
// all surviving b128 loads are issued back-to-back before any consumer.

#define BLOCK_THREADS 256
#define WAVES_PER_BLOCK 8

namespace {
constexpr int B  = 4;
constexpr int H  = 64;
constexpr int W  = 64;
constexpr int C  = 128;
constexpr int NT = 10;
constexpr float INIT_H = 256.0f;
constexpr float INIT_W = 256.0f;
}  // namespace

__global__ __launch_bounds__(BLOCK_THREADS)
void perspective_avg_kernel(const float* __restrict__ image,
                            const float* __restrict__ transforms,
                            const float* __restrict__ mask,
                            float* __restrict__ out) {
  // Per-wave slot of 16 floats (64B) for the 10 mask values.
  __shared__ float smask[WAVES_PER_BLOCK][16];

  const int lane = threadIdx.x & 31;
  const int wv   = threadIdx.x >> 5;

  // Wave-uniform pixel id; force into an SGPR so all derived address math,
  // transform loads and gate branches are scalar.
  const int pix = __builtin_amdgcn_readfirstlane((int)(blockIdx.x * WAVES_PER_BLOCK + wv));
  const int x = pix & (W - 1);
  const int y = (pix >> 6) & (H - 1);
  const int b = pix >> 12;

  // ---- Phase 0: issue the 10 strided mask loads via async global->LDS
  // (gfx1250 ASYNCcnt path). Lane n fetches mask[b, n, y, x] into its slot.
  if (lane < NT) {
    const float* gp = mask + (((size_t)(b * NT + lane) * H + (size_t)y) * W + (size_t)x);
    const uint64_t gaddr = (uint64_t)(uintptr_t)gp;
    const uint32_t laddr = (uint32_t)(uintptr_t)&smask[wv][lane];
    asm volatile("global_load_async_to_lds_b32 %0, %1, off"
                 :: "v"(laddr), "v"(gaddr)
                 : "memory");
  }

  // Homography rescale: t = transforms / mul,
  // mul = [1,1,INIT_H/H, 1,1,INIT_W/W, H/INIT_H, W/INIT_W]  (exact powers of 2)
  const float im2 = (float)H / INIT_H;   // 0.25
  const float im5 = (float)W / INIT_W;   // 0.25
  const float im6 = INIT_H / (float)H;   // 4.0
  const float im7 = INIT_W / (float)W;   // 4.0

  const float fx = (float)x;
  const float fy = (float)y;
  const float* tb    = transforms + (size_t)b * NT * 8;
  const float* img_b = image + (size_t)b * H * W * C;

  // ---- Phase A: geometry for all transforms (scalar; overlaps async loads).
  int  rowoff[NT];   // (yi*W + xi) * C, valid entries only
  bool valid [NT];
#pragma unroll
  for (int n = 0; n < NT; ++n) {
    const float t0 = tb[n * 8 + 0];
    const float t1 = tb[n * 8 + 1];
    const float t2 = tb[n * 8 + 2] * im2;
    const float t3 = tb[n * 8 + 3];
    const float t4 = tb[n * 8 + 4];
    const float t5 = tb[n * 8 + 5] * im5;
    const float t6 = tb[n * 8 + 6] * im6;
    const float t7 = tb[n * 8 + 7] * im7;

    const float denom = t6 * fx + t7 * fy + 1.0f;
    const float xin = (t0 * fx + t1 * fy + t2) / denom;
    const float yin = (t3 * fx + t4 * fy + t5) / denom;

    // NaN/inf coords fail all compares -> invalid -> contributes 0.
    const bool v = (xin >= 0.f) && (xin <= (float)(W - 1)) &&
                   (yin >= 0.f) && (yin <= (float)(H - 1));
    valid[n] = v;
    int ro = 0;
    if (v) {
      // round half-up, clip after cast (matches reference)
      int xi = (int)floorf(xin + 0.5f);
      int yi = (int)floorf(yin + 0.5f);
      xi = xi < 0 ? 0 : (xi > W - 1 ? W - 1 : xi);
      yi = yi < 0 ? 0 : (yi > H - 1 ? H - 1 : yi);
      ro = (yi * W + xi) * C;
    }
    rowoff[n] = ro;
  }

  // ---- Phase B: masks resident -> fold into scalar weights.
  asm volatile("s_wait_asynccnt 0" ::: "memory");

  float wgt[NT];
  float cnt = 0.f;
#pragma unroll
  for (int n = 0; n < NT; ++n) {
    const float m = __int_as_float(
        __builtin_amdgcn_readfirstlane(__float_as_int(smask[wv][n])));
    cnt += m;                              // count includes invalid samples
    wgt[n] = (valid[n] && (m != 0.f)) ? m : 0.f;
  }

  // ---- Phase C: issue all surviving row gathers back-to-back (pipelined).
  float4 val[NT];
#pragma unroll
  for (int n = 0; n < NT; ++n) {
    if (wgt[n] != 0.f) {                   // scalar gate; no waits in between
      const float4* src = (const float4*)(img_b + (size_t)rowoff[n]);
      val[n] = src[lane];
    } else {
      val[n] = make_float4(0.f, 0.f, 0.f, 0.f);
    }
  }

  // ---- Phase D: consume (compiler places descending loadcnt waits here).
  float4 acc = make_float4(0.f, 0.f, 0.f, 0.f);
#pragma unroll
  for (int n = 0; n < NT; ++n) {
    acc.x += wgt[n] * val[n].x;
    acc.y += wgt[n] * val[n].y;
    acc.z += wgt[n] * val[n].z;
    acc.w += wgt[n] * val[n].w;
  }

  float4 o;
  if (cnt > 0.f) {
    // Per-component true division to match jnp's s / count exactly.
    o.x = acc.x / cnt;
    o.y = acc.y / cnt;
    o.z = acc.z / cnt;
    o.w = acc.w / cnt;
  } else {
    o = make_float4(0.f, 0.f, 0.f, 0.f);
  }

  float4* op = (float4*)(out + (size_t)pix * C);
  op[lane] = o;
}

extern "C" void kernel_launch(void* const* d_in, const int* in_sizes, int n_in,
                              void* d_out, int out_size, void* d_ws, size_t ws_size,
                              hipStream_t stream) {
  (void)in_sizes; (void)n_in; (void)out_size; (void)d_ws; (void)ws_size;
  const float* image      = (const float*)d_in[0];
  const float* transforms = (const float*)d_in[1];
  const float* mask       = (const float*)d_in[2];
  float* out              = (float*)d_out;

  const int pixels = B * H * W;                     // 16384
  const int blocks = pixels / WAVES_PER_BLOCK;      // 2048
  perspective_avg_kernel<<<blocks, BLOCK_THREADS, 0, stream>>>(
      image, transforms, mask, out);
}